// MLPPredictor_45526653337663
// MI455X (gfx1250) — compile-verified
//
#include <hip/hip_runtime.h>
#include <hip/hip_bf16.h>

typedef float v2f __attribute__((ext_vector_type(2)));
typedef float v8f __attribute__((ext_vector_type(8)));

#define HFEAT   256
#define KDIM    512
#define NEDGES  300000
#define MROWS   80                      // edges per block = 5 M-tiles
#define W1S_BYTES (KDIM * HFEAT * 4)    // 512 KB swizzled-W1 workspace

// Pre-pass: W1 [512,256] row-major -> pair-interleaved so a WMMA B fragment
// (W1[k][n], W1[k+1][n]) is one contiguous b64:  W1s[(k>>1)*512 + 2n + (k&1)]
__global__ __launch_bounds__(256) void w1_swizzle_kernel(const float* __restrict__ W1,
                                                         float* __restrict__ W1s) {
    const int i = blockIdx.x * 256 + threadIdx.x;  // 131072 elements
    const int p = i >> 9;                          // k-pair index
    const int r = i & 511;                         // 2n + j
    W1s[i] = W1[(2 * p + (r & 1)) * HFEAT + (r >> 1)];
}

template <bool SWZ>
__global__ __launch_bounds__(256) void edge_mlp_wmma_kernel(
    const float* __restrict__ x,          // [N_NODES, 256]
    const int*   __restrict__ edge_index, // [2, NEDGES]
    const float* __restrict__ W1,         // [512, 256] (direct path)
    const float* __restrict__ W1s,        // pair-interleaved (SWZ path)
    const float* __restrict__ b1,
    const float* __restrict__ W2,
    const float* __restrict__ b2,
    float*       __restrict__ out)
{
    // 160 KB A tile (= half a WGP's LDS -> 2 blocks/WGP): 80 edge rows x 512 feats,
    // K index XOR-swizzled by (row&15)<<2 so the WMMA A b64 fetch covers all 64
    // banks with zero padding. Reused as the [8 wave][80 m] partial buffer after
    // the K loop.
    __shared__ float As[MROWS * KDIM];

    const int t  = threadIdx.x;
    const int e0 = blockIdx.x * MROWS;

    // ---- Stage A: row i = [ x[src[e0+i]] | x[dst[e0+i]] ], coalesced float2;
    //      edge ids fetched as int4 (1 b128 per 4 rows) ----
    {
        const int half = t >> 7;                   // 0 = src half, 1 = dst half
        const int sc   = (t & 127) << 1;           // source col in node row (even)
        const int dc   = half * HFEAT + sc;        // dest col in A row
        const int base = half * NEDGES + e0;
        #pragma unroll 2
        for (int i0 = 0; i0 < MROWS; i0 += 4) {
            const int4 nd = *(const int4*)(edge_index + base + i0);
            const int nds[4] = {nd.x, nd.y, nd.z, nd.w};
            #pragma unroll
            for (int j = 0; j < 4; ++j) {
                const int i = i0 + j;
                const float2 v = *(const float2*)(x + (long)nds[j] * HFEAT + sc);
                *(float2*)(&As[i * KDIM + (dc ^ ((i & 15) << 2))]) = v;
            }
        }
    }
    __syncthreads();

    const int lane = t & 31;
    const int wave = t >> 5;
    const int m    = lane & 15;          // row within an M-tile
    const int h    = lane >> 4;          // lane half
    const int h2   = h << 1;             // K sub-offset 0 / 2
    const int nl   = lane & 15;

    const int n0 = wave * 32 + nl;       // wave owns n-tiles 2w, 2w+1
    const int n1 = n0 + 16;
    const int swzm = m << 2;

    const float* As0 = &As[m * KDIM];          // edges e0 +  0..15
    const float* As1 = &As[(m + 16) * KDIM];   // edges e0 + 16..31
    const float* As2 = &As[(m + 32) * KDIM];   // edges e0 + 32..47
    const float* As3 = &As[(m + 48) * KDIM];   // edges e0 + 48..63
    const float* As4 = &As[(m + 64) * KDIM];   // edges e0 + 64..79

    v8f acc00 = {0.f,0.f,0.f,0.f,0.f,0.f,0.f,0.f};
    v8f acc01 = acc00, acc10 = acc00, acc11 = acc00, acc20 = acc00, acc21 = acc00;
    v8f acc30 = acc00, acc31 = acc00, acc40 = acc00, acc41 = acc00;

    // ---- [80 x 512] @ [512 x 256]: 10 WMMA per K-step (5M x 2N register tile) ----
    #pragma unroll 4
    for (int k = 0; k < KDIM; k += 4) {
        const int kc = (k + h2) ^ swzm;
        v2f a0 = *(const v2f*)(As0 + kc);
        v2f a1 = *(const v2f*)(As1 + kc);
        v2f a2 = *(const v2f*)(As2 + kc);
        v2f a3 = *(const v2f*)(As3 + kc);
        v2f a4 = *(const v2f*)(As4 + kc);

        v2f b0, b1v;
        if (SWZ) {
            const float* bp = W1s + (((k >> 1) + h) << 9);   // pair row, 512 floats
            b0  = *(const v2f*)(bp + (n0 << 1));             // one b64 per fragment
            b1v = *(const v2f*)(bp + (n1 << 1));
        } else {
            const float* w1p = W1 + (long)(k + h2) * HFEAT;
            b0.x  = w1p[n0];  b0.y  = w1p[HFEAT + n0];
            b1v.x = w1p[n1];  b1v.y = w1p[HFEAT + n1];
        }

        acc00 = __builtin_amdgcn_wmma_f32_16x16x4_f32(false, a0, false, b0,  (short)0, acc00, false, false);
        acc01 = __builtin_amdgcn_wmma_f32_16x16x4_f32(false, a0, false, b1v, (short)0, acc01, false, false);
        acc10 = __builtin_amdgcn_wmma_f32_16x16x4_f32(false, a1, false, b0,  (short)0, acc10, false, false);
        acc11 = __builtin_amdgcn_wmma_f32_16x16x4_f32(false, a1, false, b1v, (short)0, acc11, false, false);
        acc20 = __builtin_amdgcn_wmma_f32_16x16x4_f32(false, a2, false, b0,  (short)0, acc20, false, false);
        acc21 = __builtin_amdgcn_wmma_f32_16x16x4_f32(false, a2, false, b1v, (short)0, acc21, false, false);
        acc30 = __builtin_amdgcn_wmma_f32_16x16x4_f32(false, a3, false, b0,  (short)0, acc30, false, false);
        acc31 = __builtin_amdgcn_wmma_f32_16x16x4_f32(false, a3, false, b1v, (short)0, acc31, false, false);
        acc40 = __builtin_amdgcn_wmma_f32_16x16x4_f32(false, a4, false, b0,  (short)0, acc40, false, false);
        acc41 = __builtin_amdgcn_wmma_f32_16x16x4_f32(false, a4, false, b1v, (short)0, acc41, false, false);
    }

    __syncthreads();                 // A tile dead -> reuse LDS for partials
    float* partial = As;             // [8 waves][80 m]

    // ---- Fused layer 2: relu(acc + b1) * W2; both n-tiles summed BEFORE the
    //      16-lane butterfly (they feed the same output row) ----
    const float b1a = b1[n0], b1b = b1[n1];
    const float w2a = W2[n0], w2b = W2[n1];

    #pragma unroll
    for (int r = 0; r < 8; ++r) {
        float s0 = fmaxf(acc00[r] + b1a, 0.f) * w2a + fmaxf(acc01[r] + b1b, 0.f) * w2b;
        float s1 = fmaxf(acc10[r] + b1a, 0.f) * w2a + fmaxf(acc11[r] + b1b, 0.f) * w2b;
        float s2 = fmaxf(acc20[r] + b1a, 0.f) * w2a + fmaxf(acc21[r] + b1b, 0.f) * w2b;
        float s3 = fmaxf(acc30[r] + b1a, 0.f) * w2a + fmaxf(acc31[r] + b1b, 0.f) * w2b;
        float s4 = fmaxf(acc40[r] + b1a, 0.f) * w2a + fmaxf(acc41[r] + b1b, 0.f) * w2b;
        #pragma unroll
        for (int off = 8; off >= 1; off >>= 1) {   // masks <16 keep halves separate
            s0 += __shfl_xor(s0, off, 32);
            s1 += __shfl_xor(s1, off, 32);
            s2 += __shfl_xor(s2, off, 32);
            s3 += __shfl_xor(s3, off, 32);
            s4 += __shfl_xor(s4, off, 32);
        }
        if (nl == 0) {                             // lane 0 -> m=r, lane 16 -> m=r+8
            const int mr = r + (h << 3);
            partial[wave * MROWS + mr     ] = s0;
            partial[wave * MROWS + mr + 16] = s1;
            partial[wave * MROWS + mr + 32] = s2;
            partial[wave * MROWS + mr + 48] = s3;
            partial[wave * MROWS + mr + 64] = s4;
        }
    }
    __syncthreads();

    if (t < MROWS) {
        float s = b2[0];
        #pragma unroll
        for (int w = 0; w < 8; ++w) s += partial[w * MROWS + t];
        out[e0 + t] = s;
    }
}

extern "C" void kernel_launch(void* const* d_in, const int* in_sizes, int n_in,
                              void* d_out, int out_size, void* d_ws, size_t ws_size,
                              hipStream_t stream) {
    const float* x          = (const float*)d_in[0];
    const int*   edge_index = (const int*)d_in[1];
    const float* W1         = (const float*)d_in[2];
    const float* b1         = (const float*)d_in[3];
    const float* W2         = (const float*)d_in[4];
    const float* b2         = (const float*)d_in[5];
    float*       out        = (float*)d_out;

    const int nblocks = NEDGES / MROWS;   // 300000 / 80 = 3750, no tail

    if (ws_size >= (size_t)W1S_BYTES) {
        float* W1s = (float*)d_ws;
        w1_swizzle_kernel<<<(KDIM * HFEAT) / 256, 256, 0, stream>>>(W1, W1s);
        edge_mlp_wmma_kernel<true><<<nblocks, 256, 0, stream>>>(
            x, edge_index, W1, W1s, b1, W2, b2, out);
    } else {
        edge_mlp_wmma_kernel<false><<<nblocks, 256, 0, stream>>>(
            x, edge_index, W1, W1, b1, W2, b2, out);
    }
}